// AttnDecoderRNN_51264729645380
// MI455X (gfx1250) — compile-verified
//
#include <hip/hip_runtime.h>
#include <math.h>

// Problem dims (compile-time)
constexpr int BB = 64;
constexpr int SS = 2048;
constexpr int FF = 1024;
constexpr int HH = 1024;

typedef __attribute__((ext_vector_type(2))) float v2f;
typedef __attribute__((ext_vector_type(8))) float v8f;

// ---------------------------------------------------------------- utilities
__device__ __forceinline__ float wave_sum(float v) {
    #pragma unroll
    for (int off = 16; off > 0; off >>= 1) v += __shfl_xor(v, off, 32);
    return v;
}
__device__ __forceinline__ float wave_max(float v) {
    #pragma unroll
    for (int off = 16; off > 0; off >>= 1) v = fmaxf(v, __shfl_xor(v, off, 32));
    return v;
}

// ------------------------------------------------- hproj[b] = h[b].w_h + input[b]*w_i + attn_b
// grid: BB blocks x 32 threads
__global__ void hproj_kernel(const float* __restrict__ h,
                             const float* __restrict__ attn_W,
                             const float* __restrict__ attn_b,
                             const float* __restrict__ input,
                             float* __restrict__ hw) {
    const int b = blockIdx.x;
    const int lane = threadIdx.x;
    const float4* hv = (const float4*)(h + (size_t)b * HH);
    const float4* wv = (const float4*)(attn_W + FF);   // w_h section
    float acc = 0.f;
    #pragma unroll
    for (int j = 0; j < HH / 4 / 32; ++j) {
        float4 x = hv[lane + j * 32], w = wv[lane + j * 32];
        acc += x.x * w.x + x.y * w.y + x.z * w.z + x.w * w.w;
    }
    acc = wave_sum(acc);
    if (lane == 0)
        hw[b] = acc + input[b] * attn_W[FF + HH] + attn_b[0];
}

// ------------------------------------------------- scores[b,s] = E[b,s,:].w_e + hw[b] (masked)
// grid: (BB*SS/8) blocks x 256 threads (8 waves, 1 row per wave)
__global__ void scores_kernel(const float* __restrict__ E,
                              const float* __restrict__ attn_W,   // w_e = attn_W[0:FF]
                              const float* __restrict__ hw,
                              const int*   __restrict__ mask,
                              float* __restrict__ scores) {
    const int lane = threadIdx.x & 31;
    const int wv   = threadIdx.x >> 5;
    const int row  = blockIdx.x * 8 + wv;               // [0, BB*SS)
    const float4* e4 = (const float4*)(E + (size_t)row * FF);
    const float4* w4 = (const float4*)attn_W;
    float acc = 0.f;
    #pragma unroll
    for (int j = 0; j < FF / 4 / 32; ++j) {
        float4 e = e4[lane + j * 32], w = w4[lane + j * 32];
        acc += e.x * w.x + e.y * w.y + e.z * w.z + e.w * w.w;
    }
    acc = wave_sum(acc);
    if (lane == 0) {
        const int b = row >> 11;                        // row / SS
        float sc = acc + hw[b];
        if (mask[row] == 0) sc = -1.0e10f;
        scores[row] = sc;
    }
}

// ------------------------------------------------- masked softmax over S, per batch row
// grid: BB blocks x 256 threads. In-place on scores -> weights; also writes d_out section.
__global__ void softmax_kernel(float* __restrict__ scores,
                               float* __restrict__ w_dout) {
    const int b = blockIdx.x, tid = threadIdx.x;
    __shared__ float red[8];
    float* sc = scores + (size_t)b * SS;

    float mx = -3.4e38f;
    for (int s = tid; s < SS; s += 256) mx = fmaxf(mx, sc[s]);
    mx = wave_max(mx);
    if ((tid & 31) == 0) red[tid >> 5] = mx;
    __syncthreads();
    float bmax = red[0];
    #pragma unroll
    for (int i = 1; i < 8; ++i) bmax = fmaxf(bmax, red[i]);
    __syncthreads();

    float sum = 0.f;
    for (int s = tid; s < SS; s += 256) {
        float e = expf(sc[s] - bmax);
        sc[s] = e;
        sum += e;
    }
    sum = wave_sum(sum);
    if ((tid & 31) == 0) red[tid >> 5] = sum;
    __syncthreads();
    float tot = 0.f;
    #pragma unroll
    for (int i = 0; i < 8; ++i) tot += red[i];
    const float inv = 1.f / tot;

    for (int s = tid; s < SS; s += 256) {
        float w = sc[s] * inv;
        sc[s] = w;
        w_dout[(size_t)b * SS + s] = w;
    }
}

// ------------------------------------------------- attn_applied partials over S chunks
// grid: (8, BB) x 256 threads; each block: 256 rows of E, LDS-staged weights
__global__ void attn_partial_kernel(const float* __restrict__ E,
                                    const float* __restrict__ wts,
                                    float* __restrict__ part) {
    const int chunk = blockIdx.x, b = blockIdx.y, tid = threadIdx.x;
    __shared__ float wsm[256];
    wsm[tid] = wts[(size_t)b * SS + chunk * 256 + tid];
    __syncthreads();
    const float4* Eb = (const float4*)(E + ((size_t)b * SS + (size_t)chunk * 256) * FF);
    float ax = 0.f, ay = 0.f, az = 0.f, aw = 0.f;
    for (int s = 0; s < 256; ++s) {
        __builtin_prefetch(&Eb[(size_t)(s + 4) * (FF / 4) + tid], 0, 1);
        const float w = wsm[s];
        float4 e = Eb[(size_t)s * (FF / 4) + tid];
        ax += w * e.x; ay += w * e.y; az += w * e.z; aw += w * e.w;
    }
    float4 o; o.x = ax; o.y = ay; o.z = az; o.w = aw;
    ((float4*)(part + ((size_t)(b * 8 + chunk)) * FF))[tid] = o;
}

// grid: BB blocks x 256 threads — sum the 8 partials
__global__ void attn_reduce_kernel(const float* __restrict__ part,
                                   float* __restrict__ attn_applied) {
    const int b = blockIdx.x, tid = threadIdx.x;
    float ax = 0.f, ay = 0.f, az = 0.f, aw = 0.f;
    #pragma unroll
    for (int c = 0; c < 8; ++c) {
        float4 p = ((const float4*)(part + ((size_t)(b * 8 + c)) * FF))[tid];
        ax += p.x; ay += p.y; az += p.z; aw += p.w;
    }
    float4 o; o.x = ax; o.y = ay; o.z = az; o.w = aw;
    ((float4*)(attn_applied + (size_t)b * FF))[tid] = o;
}

// ------------------------------------------------- WMMA f32 GEMM:  C[m,n] = sum_k A[m,k]*W[n,k]
// Per-wave 16x16 output tile via V_WMMA_F32_16X16X4_F32. Block = 128 threads (4 waves).
// grid.x = N/64, grid.y = M/16.  Epilogue: +bias[n] (+ rank_m[m]*rank_n[n]) (+ relu)
__global__ void wmma_gemm_bt_kernel(const float* __restrict__ A, int lda,
                                    const float* __restrict__ W, int ldw,
                                    float* __restrict__ C, int ldc,
                                    const float* __restrict__ bias,
                                    const float* __restrict__ rank_m,
                                    const float* __restrict__ rank_n_base, int rank_n_stride,
                                    int K, int relu_flag) {
    const int lane  = threadIdx.x & 31;
    const int wave  = threadIdx.x >> 5;
    const int ntile = blockIdx.x * 4 + wave;
    const int mtile = blockIdx.y;
    const int r16   = lane & 15;
    const int hi2   = lane >> 4;                 // 0 or 1 (lane half)
    const float* Arow = A + (size_t)(mtile * 16 + r16) * lda;   // A fragment row
    const float* Wrow = W + (size_t)(ntile * 16 + r16) * ldw;   // B fragment column (W row)

    v8f acc = {};
    for (int k = 0; k < K; k += 4) {
        const int kk = k + hi2 * 2;
        v2f a; a.x = Arow[kk]; a.y = Arow[kk + 1];
        v2f b; b.x = Wrow[kk]; b.y = Wrow[kk + 1];
        acc = __builtin_amdgcn_wmma_f32_16x16x4_f32(
            false, a, false, b, (short)0, acc, false, false);
    }

    const int cn = ntile * 16 + r16;
    const float bn = bias ? bias[cn] : 0.f;
    const float rn = rank_m ? rank_n_base[(size_t)cn * rank_n_stride] : 0.f;
    #pragma unroll
    for (int i = 0; i < 8; ++i) {
        const int cm = mtile * 16 + i + hi2 * 8;
        float v = acc[i] + bn;
        if (rank_m) v += rank_m[cm] * rn;
        if (relu_flag) v = fmaxf(v, 0.f);
        C[(size_t)cm * ldc + cn] = v;
    }
}

// ------------------------------------------------- fused GRU gates -> h_new
// grid: BB*HH/256 blocks x 256 threads
__global__ void gru_kernel(const float* __restrict__ gi,
                           const float* __restrict__ gh,
                           const float* __restrict__ h,
                           float* __restrict__ h_new) {
    const int idx = blockIdx.x * 256 + threadIdx.x;     // [0, BB*HH)
    const int b = idx >> 10, j = idx & (HH - 1);
    const size_t o = (size_t)b * 3 * HH;
    const float r = 1.f / (1.f + expf(-(gi[o + j] + gh[o + j])));
    const float z = 1.f / (1.f + expf(-(gi[o + HH + j] + gh[o + HH + j])));
    const float n = tanhf(gi[o + 2 * HH + j] + r * gh[o + 2 * HH + j]);
    h_new[idx] = (1.f - z) * n + z * h[idx];
}

// ------------------------------------------------- out[b] = relu(h_new[b,:]).out_W + out_b
// grid: BB blocks x 32 threads
__global__ void out_kernel(const float* __restrict__ h_new,
                           const float* __restrict__ out_W,
                           const float* __restrict__ out_b,
                           float* __restrict__ out) {
    const int b = blockIdx.x, lane = threadIdx.x;
    const float4* hv = (const float4*)(h_new + (size_t)b * HH);
    const float4* wv = (const float4*)out_W;
    float acc = 0.f;
    #pragma unroll
    for (int j = 0; j < HH / 4 / 32; ++j) {
        float4 x = hv[lane + j * 32], w = wv[lane + j * 32];
        acc += fmaxf(x.x, 0.f) * w.x + fmaxf(x.y, 0.f) * w.y +
               fmaxf(x.z, 0.f) * w.z + fmaxf(x.w, 0.f) * w.w;
    }
    acc = wave_sum(acc);
    if (lane == 0) out[b] = acc + out_b[0];
}

// ================================================================ launch
extern "C" void kernel_launch(void* const* d_in, const int* in_sizes, int n_in,
                              void* d_out, int out_size, void* d_ws, size_t ws_size,
                              hipStream_t stream) {
    const float* input     = (const float*)d_in[0];    // (B,)
    const float* hidden    = (const float*)d_in[1];    // (1,B,H) -> B*H
    const float* enc       = (const float*)d_in[2];    // (B,S,F)
    const int*   mask      = (const int*)  d_in[3];    // (B,S)
    const float* attn_W    = (const float*)d_in[4];    // (1, F+H+1)
    const float* attn_b    = (const float*)d_in[5];    // (1,)
    const float* combine_W = (const float*)d_in[6];    // (H, F+1)
    const float* combine_b = (const float*)d_in[7];    // (H,)
    const float* W_ih      = (const float*)d_in[8];    // (3H, H)
    const float* W_hh      = (const float*)d_in[9];    // (3H, H)
    const float* b_ih      = (const float*)d_in[10];   // (3H,)
    const float* b_hh      = (const float*)d_in[11];   // (3H,)
    const float* out_W     = (const float*)d_in[12];   // (1, H)
    const float* out_b     = (const float*)d_in[13];   // (1,)

    // d_out layout: [0,64) out ; [64, 64+B*H) h_new ; then attn_weights (B*S)
    float* o_out   = (float*)d_out;
    float* o_hnew  = o_out + BB;
    float* o_attnw = o_hnew + (size_t)BB * HH;

    // workspace layout (floats)
    float* w  = (float*)d_ws;
    float* hw           = w;                 w += BB;                    // 64
    float* scores       = w;                 w += (size_t)BB * SS;       // 131072 (-> weights in place)
    float* part         = w;                 w += (size_t)BB * 8 * FF;   // 524288
    float* attn_applied = w;                 w += (size_t)BB * FF;       // 65536
    float* xbuf         = w;                 w += (size_t)BB * HH;       // 65536
    float* gi           = w;                 w += (size_t)BB * 3 * HH;   // 196608
    float* gh           = w;                 w += (size_t)BB * 3 * HH;   // 196608

    // 1) h-projection scalar per batch row
    hproj_kernel<<<BB, 32, 0, stream>>>(hidden, attn_W, attn_b, input, hw);

    // 2) attention scores (streams 512MB of encoder_outputs)
    scores_kernel<<<(BB * SS) / 8, 256, 0, stream>>>(enc, attn_W, hw, mask, scores);

    // 3) masked softmax; weights stay in `scores` and go to d_out
    softmax_kernel<<<BB, 256, 0, stream>>>(scores, o_attnw);

    // 4) attn_applied = weights . encoder_outputs (second 512MB stream)
    attn_partial_kernel<<<dim3(8, BB), 256, 0, stream>>>(enc, scores, part);
    attn_reduce_kernel<<<BB, 256, 0, stream>>>(part, attn_applied);

    // 5) combine layer: x = relu([input, attn_applied] @ combine_W^T + combine_b)
    //    K=1024 WMMA GEMM over attn_applied vs combine_W[:,1:], rank-1 term for input column.
    wmma_gemm_bt_kernel<<<dim3(HH / 64, BB / 16), 128, 0, stream>>>(
        attn_applied, FF,
        combine_W + 1, FF + 1,
        xbuf, HH,
        combine_b,
        input, combine_W, FF + 1,
        FF, /*relu=*/1);

    // 6) GRU input/hidden projections (M=64, N=3072, K=1024)
    wmma_gemm_bt_kernel<<<dim3(3 * HH / 64, BB / 16), 128, 0, stream>>>(
        xbuf, HH, W_ih, HH, gi, 3 * HH, b_ih,
        nullptr, nullptr, 0, HH, 0);
    wmma_gemm_bt_kernel<<<dim3(3 * HH / 64, BB / 16), 128, 0, stream>>>(
        hidden, HH, W_hh, HH, gh, 3 * HH, b_hh,
        nullptr, nullptr, 0, HH, 0);

    // 7) fused gates -> h_new (written straight into d_out)
    gru_kernel<<<(BB * HH) / 256, 256, 0, stream>>>(gi, gh, hidden, o_hnew);

    // 8) output projection
    out_kernel<<<BB, 32, 0, stream>>>(o_hnew, out_W, out_b, o_out);
}